// Linear_14577119003293
// MI455X (gfx1250) — compile-verified
//
#include <hip/hip_runtime.h>

// CDNA5 (gfx1250) wave32 WMMA f32 path.
typedef float v2f __attribute__((ext_vector_type(2)));
typedef float v4f __attribute__((ext_vector_type(4)));
typedef float v8f __attribute__((ext_vector_type(8)));

#define BATCH 256
#define IN_F  64
#define OUT_F 64
#define LWE   4096

// Each wave computes a full M=64 (all OUT_F) x N=16 (d columns) strip for one
// batch, accumulating K=IN_F=64 in steps of 4 via V_WMMA_F32_16X16X4_F32.
// X is streamed from HBM exactly once; W is staged in LDS per block.
__global__ __launch_bounds__(256) void lwe_linear_wmma(
    const float* __restrict__ X,     // [BATCH, IN_F, LWE]
    const float* __restrict__ W,     // [OUT_F, IN_F]
    const float* __restrict__ bias,  // [OUT_F]
    float* __restrict__ out)         // [BATCH, OUT_F, LWE]
{
    __shared__ float Ws[OUT_F * IN_F];  // 16 KB

    // Stage W row-major into LDS: 256 threads x 4 float4 = 4096 floats.
    {
        const v4f* Wg = reinterpret_cast<const v4f*>(W);
        v4f*       Wl = reinterpret_cast<v4f*>(Ws);
        const int  t  = threadIdx.x;
#pragma unroll
        for (int k = 0; k < 4; ++k)
            Wl[t + 256 * k] = Wg[t + 256 * k];
    }
    __syncthreads();

    const int lane = threadIdx.x & 31;
    const int wave = threadIdx.x >> 5;
    const int tile = blockIdx.x * 8 + wave;  // 65536 wave-tiles total
    const int b    = tile >> 8;              // batch index (0..255)
    const int d0   = (tile & 255) << 4;      // d column base, 16 wide

    const int hi = lane >> 4;  // 0 = lanes 0..15, 1 = lanes 16..31
    const int lo = lane & 15;

    // B-matrix (4x16, K x N) base address for this lane:
    //   V0 holds rows K={0,1} (split by lane half), V1 holds rows K={2,3}.
    const float* xb = X + (size_t)b * IN_F * LWE + (size_t)hi * LWE + d0 + lo;

    v8f c[4] = {};  // 4 M-tiles of 16x16 f32 accumulators (32 VGPRs)

#pragma unroll
    for (int kk = 0; kk < 16; ++kk) {
        // Load B 4x16 tile: two coalesced b32 loads (rows hi and hi+2).
        v2f bv;
        bv.x = xb[(size_t)(4 * kk + 0) * LWE];
        bv.y = xb[(size_t)(4 * kk + 2) * LWE];

#pragma unroll
        for (int mt = 0; mt < 4; ++mt) {
            // A 16x4 tile from LDS in ISA layout:
            //   lanes 0-15:  (V0,V1) = W[o, 4kk+0], W[o, 4kk+1]
            //   lanes 16-31: (V0,V1) = W[o, 4kk+2], W[o, 4kk+3]
            // with o = 16*mt + lo.  Consecutive in row-major W -> one b64.
            const v2f a = *reinterpret_cast<const v2f*>(
                &Ws[(16 * mt + lo) * IN_F + 4 * kk + 2 * hi]);
            c[mt] = __builtin_amdgcn_wmma_f32_16x16x4_f32(
                /*neg_a=*/false, a, /*neg_b=*/false, bv,
                /*c_mod=*/(short)0, c[mt],
                /*reuse_a=*/false, /*reuse_b=*/false);
        }
    }

    // Store. C/D layout: VGPR r, lanes 0-15 -> M = r, lanes 16-31 -> M = r+8.
    // Element (mt, r, lane): o = 16*mt + r + 8*hi, d = d0 + lo.
    const int d = d0 + lo;
    float* ob = out + (size_t)b * OUT_F * LWE + d0 + lo;
#pragma unroll
    for (int mt = 0; mt < 4; ++mt) {
#pragma unroll
        for (int r = 0; r < 8; ++r) {
            const int o = 16 * mt + r + 8 * hi;
            float v = c[mt][r];
            if (d == LWE - 1) v += bias[o];  // body shift, single lane
            ob[(size_t)o * LWE] = v;
        }
    }
}

extern "C" void kernel_launch(void* const* d_in, const int* in_sizes, int n_in,
                              void* d_out, int out_size, void* d_ws, size_t ws_size,
                              hipStream_t stream) {
    (void)in_sizes; (void)n_in; (void)out_size; (void)d_ws; (void)ws_size;
    const float* X    = (const float*)d_in[0];  // input  [256,64,4096]
    const float* W    = (const float*)d_in[1];  // weight [64,64]
    const float* bias = (const float*)d_in[2];  // bias   [64]
    float* out        = (float*)d_out;          // [256,64,4096]

    // 65536 wave-tiles (256 batches x 256 d-chunks), 8 waves per block.
    dim3 grid(8192), block(256);
    hipLaunchKernelGGL(lwe_linear_wmma, grid, block, 0, stream, X, W, bias, out);
}